// InfoNCELoss_29111288332443
// MI455X (gfx1250) — compile-verified
//
#include <hip/hip_runtime.h>
#include <math.h>
#include <stdint.h>

// ---- CDNA5 WMMA vector types -------------------------------------------------
typedef __attribute__((ext_vector_type(2))) float v2f;   // A/B frag of 16x16x4 f32
typedef __attribute__((ext_vector_type(8))) float v8f;   // 16x16 f32 C/D frag
typedef __attribute__((ext_vector_type(4))) unsigned int u32x4;
typedef __attribute__((ext_vector_type(8))) unsigned int u32x8;

#define N_ROWS   4096
#define DIMK     512
#define TEMP_INV 10.0f      // 1 / 0.1
#define EPS_NRM  1e-8f
#define NEG_BIG  -1e30f
#define LDA      (DIMK + 4) // 516-float LDS row stride (TDM pad: +2 dw per 256 dw)

// ============================================================================
// TDM: DMA a [rows x 512] f32 tile from global into LDS via the Tensor Data
// Mover (CDNA5 ISA). Padding config inserts 2 DWORDs
// after every 256 DWORDs -> effective LDS row stride 516 floats (bank-conflict
// free for the WMMA fragment reads), with the K>=256 half shifted by +2.
// ============================================================================
__device__ __forceinline__ void tdm_load_tile_f32(const float* gptr,
                                                  unsigned lds_byte_off,
                                                  unsigned rows) {
    const unsigned long long ga = (unsigned long long)(uintptr_t)gptr;

    u32x4 g0;
    g0[0] = 1u;                                        // count=1, user D#
    g0[1] = lds_byte_off;                              // LDS byte address
    g0[2] = (unsigned)(ga & 0xffffffffu);              // global_addr[31:0]
    g0[3] = (unsigned)((ga >> 32) & 0x01ffffffu)       // global_addr[56:32]
          | (2u << 30);                                // type = 2 ("image")

    u32x8 g1;
    g1[0] = (2u << 16)                                 // data_size = 4 bytes
          | (1u << 20)                                 // pad_enable
          | (7u << 22)                                 // pad_interval: 256 dw
          | (1u << 25);                                // pad_amount:   2 dw
    g1[1] = ((unsigned)DIMK & 0xffffu) << 16;          // tensor_dim0[15:0] = 512
    g1[2] = (rows & 0xffffu) << 16;                    // tensor_dim1[15:0] = rows
    g1[3] = ((unsigned)DIMK & 0xffffu) << 16;          // tile_dim0 = 512
    g1[4] = rows & 0xffffu;                            // tile_dim1 = rows
    g1[5] = (unsigned)DIMK;                            // tensor_dim0_stride = 512
    g1[6] = 0u;
    g1[7] = 0u;

    // 2-D tensor: descriptor groups 2/3 disabled (VADDR2/3 = NULL)
    asm volatile("tensor_load_to_lds %0, %1" :: "s"(g0), "s"(g1) : "memory");
}

__device__ __forceinline__ void tdm_wait0() {
    asm volatile("s_wait_tensorcnt 0x0" ::: "memory");
}

// ============================================================================
// Kernel 1: per-row inverse L2 norm (clamped):  invn[i] = 1 / max(||x_i||, eps)
// ============================================================================
__global__ void k_invnorm(const float* __restrict__ F, float* __restrict__ invn) {
    __shared__ float red[4];
    const int row = blockIdx.x;
    const int t   = threadIdx.x;                    // 0..127
    float4 v = *(const float4*)(F + (size_t)row * DIMK + t * 4);
    float ss = v.x * v.x + v.y * v.y + v.z * v.z + v.w * v.w;
    for (int off = 16; off > 0; off >>= 1)
        ss += __shfl_down(ss, off, 32);
    if ((t & 31) == 0) red[t >> 5] = ss;
    __syncthreads();
    if (t == 0) {
        float s = red[0] + red[1] + red[2] + red[3];
        invn[row] = 1.0f / fmaxf(sqrtf(s), EPS_NRM);
    }
}

// ============================================================================
// Kernel 2: fused Gram-matrix + online softmax (flash style).
// 64 rows per WG (4 wave32); A block TDM'd into LDS once, each 16-col B tile
// TDM'd per iteration; each wave accumulates a 16x16 fp32 tile with 128
// chained v_wmma_f32_16x16x4_f32, then updates running (m, s) per row.
// ============================================================================
__global__ void k_sim_softmax(const float* __restrict__ F,
                              const float* __restrict__ invn,
                              float* __restrict__ row_m,
                              float* __restrict__ row_s) {
    extern __shared__ float smem[];
    float* As = smem;                         // 64 x 516
    float* Bs = As + 64 * LDA;                // 16 x 516
    float* Ms = Bs + 16 * LDA;                // 64 running max
    float* Ss = Ms + 64;                      // 64 running sum

    const int tid  = threadIdx.x;             // 0..127
    const int lane = tid & 31;
    const int wave = tid >> 5;                // 0..3
    const int row0 = blockIdx.x * 64;

    const unsigned as_off = (unsigned)(uintptr_t)(void*)As;  // LDS byte addr
    const unsigned bs_off = (unsigned)(uintptr_t)(void*)Bs;

    // ---- TDM: stage resident A block (64 rows x 512 f32) ---------------------
    if (wave == 0) {
        tdm_load_tile_f32(F + (size_t)row0 * DIMK, as_off, 64);
        tdm_wait0();
    }
    if (tid < 64) { Ms[tid] = NEG_BIG; Ss[tid] = 0.0f; }

    // ---- WMMA fragment geometry (ISA 7.12.2, 32-bit A 16x4 / B 4x16 / C 16x16)
    const int half  = lane >> 4;              // 0: K+0/K+1   1: K+2/K+3
    const int nlane = lane & 15;              // M (A) / N (B,C)
    const int kd    = half * 2;
    const int lrow  = wave * 16 + nlane;      // A local row for this lane

    float rowinv[8];                          // 10 * inv_norm for my 8 C rows
    #pragma unroll
    for (int r = 0; r < 8; ++r)
        rowinv[r] = invn[row0 + wave * 16 + r + half * 8] * TEMP_INV;

    for (int ct = 0; ct < N_ROWS / 16; ++ct) {
        const int c0 = ct * 16;
        __syncthreads();                      // previous tile done reading Bs
        if (wave == 0) {                      // TDM: stage B tile (16 x 512 f32)
            tdm_load_tile_f32(F + (size_t)c0 * DIMK, bs_off, 16);
            tdm_wait0();
        }
        __syncthreads();

        // ---- 16x16 fp32 tile: K=512 in steps of 4; TDM pad shifts the
        //      K>=256 half of each row by +2 floats.
        v8f acc = {};
        #pragma unroll
        for (int h = 0; h < 2; ++h) {
            const float* ap = As + lrow  * LDA + kd + h * (256 + 2);
            const float* bp = Bs + nlane * LDA + kd + h * (256 + 2);
            #pragma unroll 8
            for (int k = 0; k < 256; k += 4) {
                v2f a = *(const v2f*)(ap + k);   // ds_load_b64: {K, K+1} (+kd)
                v2f b = *(const v2f*)(bp + k);
                acc = __builtin_amdgcn_wmma_f32_16x16x4_f32(
                          /*neg_a=*/false, a, /*neg_b=*/false, b,
                          /*c_mod=*/(short)0, acc,
                          /*reuse_a=*/false, /*reuse_b=*/false);
            }
        }

        // ---- online softmax update; C layout: VGPR r -> row r (lanes<16) / r+8
        const float cinv = invn[c0 + nlane];
        const int   gcol = c0 + nlane;
        #pragma unroll
        for (int r = 0; r < 8; ++r) {
            float x = acc[r] * (rowinv[r] * cinv);          // logits = 10*cos_sim
            const int grow = row0 + wave * 16 + r + half * 8;
            if (grow == gcol) x = NEG_BIG;                  // drop diagonal
            float mt = x;                                   // row max over 16 cols
            for (int off = 1; off < 16; off <<= 1)
                mt = fmaxf(mt, __shfl_xor(mt, off, 16));
            const int lr = wave * 16 + r + half * 8;
            const float om = Ms[lr], os = Ss[lr];
            const float nm = fmaxf(om, mt);
            float ps = expf(x - nm);                        // row sum of exp
            for (int off = 1; off < 16; off <<= 1)
                ps += __shfl_xor(ps, off, 16);
            const float ns = os * expf(om - nm) + ps;
            if (nlane == 0) { Ms[lr] = nm; Ss[lr] = ns; }   // wave-coherent LDS
        }
    }
    __syncthreads();
    if (tid < 64) { row_m[row0 + tid] = Ms[tid]; row_s[row0 + tid] = Ss[tid]; }
}

// ============================================================================
// Kernel 3: per-row target + loss. target = first j != i with matching label
// (reference argmax-of-first-positive); fallback to column 0 of the N-1 layout.
// ============================================================================
__global__ void k_row_loss(const float* __restrict__ F,
                           const int* __restrict__ labels,
                           const float* __restrict__ invn,
                           const float* __restrict__ row_m,
                           const float* __restrict__ row_s,
                           float* __restrict__ loss) {
    __shared__ int   imin[128];
    __shared__ float fred[128];
    const int i  = blockIdx.x;
    const int t  = threadIdx.x;               // 0..127
    const int li = labels[i];

    int best = 0x7fffffff;
    for (int j = t; j < N_ROWS; j += 128)
        if (j != i && labels[j] == li) best = min(best, j);
    imin[t] = best;
    __syncthreads();
    for (int off = 64; off > 0; off >>= 1) {
        if (t < off) imin[t] = min(imin[t], imin[t + off]);
        __syncthreads();
    }
    int tgt = imin[0];
    if (tgt == 0x7fffffff) tgt = (i == 0) ? 1 : 0;  // no positive: column 0

    float4 a = *(const float4*)(F + (size_t)i   * DIMK + t * 4);
    float4 b = *(const float4*)(F + (size_t)tgt * DIMK + t * 4);
    fred[t] = a.x * b.x + a.y * b.y + a.z * b.z + a.w * b.w;
    __syncthreads();
    for (int off = 64; off > 0; off >>= 1) {
        if (t < off) fred[t] += fred[t + off];
        __syncthreads();
    }
    if (t == 0) {
        const float sim = fred[0] * invn[i] * invn[tgt];
        const float lse = row_m[i] + logf(row_s[i]);        // logsumexp_{j!=i}
        loss[i] = lse - TEMP_INV * sim;                     // -logp[target]
    }
}

// ============================================================================
// Kernel 4: mean over rows -> scalar (single block, deterministic, no atomics)
// ============================================================================
__global__ void k_reduce(const float* __restrict__ loss, float* __restrict__ out) {
    __shared__ float red[256];
    const int t = threadIdx.x;
    float s = 0.0f;
    for (int j = t; j < N_ROWS; j += 256) s += loss[j];
    red[t] = s;
    __syncthreads();
    for (int off = 128; off > 0; off >>= 1) {
        if (t < off) red[t] += red[t + off];
        __syncthreads();
    }
    if (t == 0) out[0] = red[0] / (float)N_ROWS;
}

// ============================================================================
extern "C" void kernel_launch(void* const* d_in, const int* in_sizes, int n_in,
                              void* d_out, int out_size, void* d_ws, size_t ws_size,
                              hipStream_t stream) {
    const float* F      = (const float*)d_in[0];   // (4096, 512) fp32
    const int*   labels = (const int*)d_in[1];     // (4096,) int
    float*       out    = (float*)d_out;           // scalar fp32 loss
    float*       ws     = (float*)d_ws;

    float* invn  = ws;                 // 4096
    float* rm    = ws + 1 * N_ROWS;    // 4096
    float* rs    = ws + 2 * N_ROWS;    // 4096
    float* loss  = ws + 3 * N_ROWS;    // 4096   (64 KB total scratch)

    k_invnorm<<<N_ROWS, 128, 0, stream>>>(F, invn);

    const size_t smem = (size_t)(64 + 16) * LDA * sizeof(float)
                      + 2 * 64 * sizeof(float);   // ~161.8 KB dynamic LDS (<=320KB/WGP)
    k_sim_softmax<<<N_ROWS / 64, 128, smem, stream>>>(F, invn, rm, rs);

    k_row_loss<<<N_ROWS, 128, 0, stream>>>(F, labels, invn, rm, rs, loss);
    k_reduce<<<1, 256, 0, stream>>>(loss, out);
}